// MultiHeadAttention_4389456577178
// MI455X (gfx1250) — compile-verified
//
#include <hip/hip_runtime.h>
#include <hip/hip_bf16.h>
#include <math.h>

typedef __attribute__((ext_vector_type(2))) float v2f;
typedef __attribute__((ext_vector_type(8))) float v8f;

#define CDIM 128
#define C3   384
#define HH   128
#define WW   128
#define PP   (HH * WW)      // 16384 pixels
#define LP   132            // LDS row pitch (floats) to avoid bank conflicts

// D = A(16x4) * B(4x16) + C, all f32, one wave.
__device__ __forceinline__ v8f wmma4(v2f a, v2f b, v8f c) {
  return __builtin_amdgcn_wmma_f32_16x16x4_f32(
      /*neg_a=*/false, a, /*neg_b=*/false, b,
      /*c_mod=*/(short)0, c, /*reuse_a=*/false, /*reuse_b=*/false);
}

// ---------------- Weight repack: src[O][I][S] (OIHW) -> dst[S][I][O]
// Makes the WMMA A-operand load lane-contiguous in the output-channel dim.
__global__ __launch_bounds__(256) void repack_kernel(const float* __restrict__ src,
                                                     float* __restrict__ dst,
                                                     int O, int I, int S) {
  const int idx = blockIdx.x * 256 + threadIdx.x;
  const int total = O * I * S;
  if (idx >= total) return;
  const int s = idx % S;
  const int i = (idx / S) % I;
  const int o = idx / (S * I);
  dst[((size_t)s * I + i) * O + o] = src[idx];
}

// ---------------- Stage 1: 1x1 conv (qkv1), 32(M) x 16(N) tile per wave.
// out1[b][oc][p] = sum_ic W1[oc][ic] * x[b][ic][p];  w1t layout [ic][oc].
__global__ __launch_bounds__(32) void qkv1_kernel(const float* __restrict__ x,
                                                  const float* __restrict__ w1t,
                                                  float* __restrict__ out1) {
  const int n0 = blockIdx.x * 16;
  const int m0 = blockIdx.y * 32;
  const int b  = blockIdx.z;
  const int lane = threadIdx.x;
  const int l = lane & 15, half = lane >> 4;
  const float* xb = x + (size_t)b * CDIM * PP;
  const int n = n0 + l;
  v8f acc0 = {}, acc1 = {};
  for (int k0 = 0; k0 < CDIM; k0 += 4) {
    const int ks = k0 + half * 2;
    v2f a0, a1, bb;
    a0.x = w1t[(size_t)ks * C3 + m0 + l];
    a0.y = w1t[(size_t)(ks + 1) * C3 + m0 + l];
    a1.x = w1t[(size_t)ks * C3 + m0 + 16 + l];
    a1.y = w1t[(size_t)(ks + 1) * C3 + m0 + 16 + l];
    bb.x = xb[(size_t)ks * PP + n];
    bb.y = xb[(size_t)(ks + 1) * PP + n];
    acc0 = wmma4(a0, bb, acc0);
    acc1 = wmma4(a1, bb, acc1);
  }
  float* ob = out1 + (size_t)b * C3 * PP;
#pragma unroll
  for (int r = 0; r < 8; ++r) {
    ob[(size_t)(m0 + half * 8 + r) * PP + n]      = acc0[r];
    ob[(size_t)(m0 + 16 + half * 8 + r) * PP + n] = acc1[r];
  }
}

// ---------------- Stage 2: 3x3 SAME conv (qkv2), implicit GEMM, K = 384*9.
// 32(M) x 32(N) register block per wave: 8 loads per 4 WMMAs.
// wp layout: [kh*3+kw][ic][oc].
__global__ __launch_bounds__(32) void qkv2_kernel(const float* __restrict__ in1,
                                                  const float* __restrict__ wp,
                                                  float* __restrict__ out2) {
  const int pt = blockIdx.x;                 // 512 tiles: 4 per row, 32 wide
  const int h  = pt >> 2;
  const int w0 = (pt & 3) * 32;
  const int m0 = blockIdx.y * 32;
  const int b  = blockIdx.z;
  const int lane = threadIdx.x;
  const int l = lane & 15, half = lane >> 4;
  const float* ib = in1 + (size_t)b * C3 * PP;
  v8f acc00 = {}, acc01 = {}, acc10 = {}, acc11 = {};
  for (int kh = 0; kh < 3; ++kh) {
    const int hh = h + kh - 1;               // wave-uniform
    if (hh < 0 || hh >= HH) continue;
    for (int kw = 0; kw < 3; ++kw) {
      const float* wt = wp + (size_t)(kh * 3 + kw) * C3 * C3;
      const int ww0 = w0 + l + kw - 1;
      const int ww1 = ww0 + 16;
      const bool ok0 = (ww0 >= 0) && (ww0 < WW);   // lane predicates, no branch
      const bool ok1 = (ww1 >= 0) && (ww1 < WW);
      const float* col0 = ib + (size_t)hh * WW + (ok0 ? ww0 : 0);
      const float* col1 = ib + (size_t)hh * WW + (ok1 ? ww1 : 0);
      for (int ic0 = 0; ic0 < C3; ic0 += 4) {
        const int ks = ic0 + half * 2;
        v2f a0, a1, b0, b1;
        a0.x = wt[(size_t)ks * C3 + m0 + l];
        a0.y = wt[(size_t)(ks + 1) * C3 + m0 + l];
        a1.x = wt[(size_t)ks * C3 + m0 + 16 + l];
        a1.y = wt[(size_t)(ks + 1) * C3 + m0 + 16 + l];
        b0.x = ok0 ? col0[(size_t)ks * PP] : 0.0f;
        b0.y = ok0 ? col0[(size_t)(ks + 1) * PP] : 0.0f;
        b1.x = ok1 ? col1[(size_t)ks * PP] : 0.0f;
        b1.y = ok1 ? col1[(size_t)(ks + 1) * PP] : 0.0f;
        acc00 = wmma4(a0, b0, acc00);
        acc01 = wmma4(a0, b1, acc01);
        acc10 = wmma4(a1, b0, acc10);
        acc11 = wmma4(a1, b1, acc11);
      }
    }
  }
  float* ob = out2 + (size_t)b * C3 * PP + (size_t)h * WW;
#pragma unroll
  for (int r = 0; r < 8; ++r) {
    ob[(size_t)(m0 + half * 8 + r) * PP + w0 + l]           = acc00[r];
    ob[(size_t)(m0 + half * 8 + r) * PP + w0 + 16 + l]      = acc01[r];
    ob[(size_t)(m0 + 16 + half * 8 + r) * PP + w0 + l]      = acc10[r];
    ob[(size_t)(m0 + 16 + half * 8 + r) * PP + w0 + 16 + l] = acc11[r];
  }
}

// ---------------- Stage 3: per-(b,c) channel attention over 128x128 spatial slices
// qn = l2norm rows(q), kn = l2norm rows(k); S = qn*kn^T * scale[c]; P = softmax(S);
// out = P * v.   One 256-thread (8-wave) workgroup per (b,c).
__global__ __launch_bounds__(256) void attn_kernel(const float* __restrict__ qkv,
                                                   const float* __restrict__ scale,
                                                   float* __restrict__ out) {
  extern __shared__ float smem[];
  float* qn = smem;                    // 128 x LP
  float* kn = smem + 128 * LP;         // 128 x LP
  float* S  = smem + 2 * 128 * LP;     // 128 x LP
  const int c = blockIdx.x;
  const int b = blockIdx.y;
  const float* qb = qkv + ((size_t)b * C3 + c) * PP;
  const float* kb = qkv + ((size_t)b * C3 + 128 + c) * PP;
  const float* vb = qkv + ((size_t)b * C3 + 256 + c) * PP;
  const int tid = threadIdx.x;

  // Load + L2-normalize rows: threads 0-127 -> q rows, 128-255 -> k rows.
  {
    const int r = tid & 127;
    const float* src = (tid < 128) ? qb : kb;
    float* dst = (tid < 128) ? qn : kn;
    float ss = 0.0f;
    for (int j = 0; j < WW; ++j) {
      const float v = src[(size_t)r * WW + j];
      dst[r * LP + j] = v;
      ss += v * v;
    }
    const float inv = 1.0f / fmaxf(sqrtf(ss), 1e-12f);
    for (int j = 0; j < WW; ++j) dst[r * LP + j] *= inv;
  }
  __syncthreads();

  const int wv = tid >> 5;
  const int lane = tid & 31;
  const int l = lane & 15, half = lane >> 4;

  // S = qn * kn^T   (K = spatial w)
  for (int ti = wv; ti < 64; ti += 8) {
    const int m0 = (ti >> 3) * 16, n0 = (ti & 7) * 16;
    v8f acc = {};
    for (int k0 = 0; k0 < WW; k0 += 4) {
      const int ks = k0 + half * 2;
      v2f a, bb;
      a.x = qn[(m0 + l) * LP + ks];
      a.y = qn[(m0 + l) * LP + ks + 1];
      bb.x = kn[(n0 + l) * LP + ks];    // B[k][n] = kn[n][k]
      bb.y = kn[(n0 + l) * LP + ks + 1];
      acc = wmma4(a, bb, acc);
    }
#pragma unroll
    for (int r = 0; r < 8; ++r)
      S[(m0 + half * 8 + r) * LP + n0 + l] = acc[r];
  }
  __syncthreads();

  // softmax over last dim with per-channel scale
  if (tid < 128) {
    const float sc = scale[c];
    float* row = S + tid * LP;
    float mx = -INFINITY;
    for (int j = 0; j < 128; ++j) { row[j] *= sc; mx = fmaxf(mx, row[j]); }
    float sum = 0.0f;
    for (int j = 0; j < 128; ++j) { const float e = expf(row[j] - mx); row[j] = e; sum += e; }
    const float inv = 1.0f / sum;
    for (int j = 0; j < 128; ++j) row[j] *= inv;
  }
  __syncthreads();

  // out = P * v   (K = spatial g), v read straight from global (L2-hot)
  float* ob = out + ((size_t)b * CDIM + c) * PP;
  for (int ti = wv; ti < 64; ti += 8) {
    const int m0 = (ti >> 3) * 16, n0 = (ti & 7) * 16;
    v8f acc = {};
    for (int k0 = 0; k0 < HH; k0 += 4) {
      const int ks = k0 + half * 2;
      v2f a, bb;
      a.x = S[(m0 + l) * LP + ks];
      a.y = S[(m0 + l) * LP + ks + 1];
      bb.x = vb[(size_t)ks * WW + n0 + l];
      bb.y = vb[(size_t)(ks + 1) * WW + n0 + l];
      acc = wmma4(a, bb, acc);
    }
#pragma unroll
    for (int r = 0; r < 8; ++r)
      ob[(size_t)(m0 + half * 8 + r) * WW + n0 + l] = acc[r];
  }
}

// ---------------- Stage 4: 1x1 conv (proj), 32(M) x 16(N) tile per wave.
// projt layout [ic][oc].
__global__ __launch_bounds__(32) void proj_kernel(const float* __restrict__ in,
                                                  const float* __restrict__ wpt,
                                                  float* __restrict__ out) {
  const int n0 = blockIdx.x * 16;
  const int m0 = blockIdx.y * 32;
  const int b  = blockIdx.z;
  const int lane = threadIdx.x;
  const int l = lane & 15, half = lane >> 4;
  const float* ib = in + (size_t)b * CDIM * PP;
  const int n = n0 + l;
  v8f acc0 = {}, acc1 = {};
  for (int k0 = 0; k0 < CDIM; k0 += 4) {
    const int ks = k0 + half * 2;
    v2f a0, a1, bb;
    a0.x = wpt[(size_t)ks * CDIM + m0 + l];
    a0.y = wpt[(size_t)(ks + 1) * CDIM + m0 + l];
    a1.x = wpt[(size_t)ks * CDIM + m0 + 16 + l];
    a1.y = wpt[(size_t)(ks + 1) * CDIM + m0 + 16 + l];
    bb.x = ib[(size_t)ks * PP + n];
    bb.y = ib[(size_t)(ks + 1) * PP + n];
    acc0 = wmma4(a0, bb, acc0);
    acc1 = wmma4(a1, bb, acc1);
  }
  float* ob = out + (size_t)b * CDIM * PP;
#pragma unroll
  for (int r = 0; r < 8; ++r) {
    ob[(size_t)(m0 + half * 8 + r) * PP + n]      = acc0[r];
    ob[(size_t)(m0 + 16 + half * 8 + r) * PP + n] = acc1[r];
  }
}

extern "C" void kernel_launch(void* const* d_in, const int* in_sizes, int n_in,
                              void* d_out, int out_size, void* d_ws, size_t ws_size,
                              hipStream_t stream) {
  const float* x      = (const float*)d_in[0];   // (4,128,128,128)
  const float* qkv1_w = (const float*)d_in[1];   // (384,128,1,1)
  const float* qkv2_w = (const float*)d_in[2];   // (384,384,3,3)
  const float* proj_w = (const float*)d_in[3];   // (128,128,1,1)
  const float* scale  = (const float*)d_in[4];   // (128,1,1)
  float* out = (float*)d_out;                    // (4,128,128,128)

  const size_t qkvElems = (size_t)4 * C3 * PP;   // 25,165,824 floats
  float* ws1 = (float*)d_ws;                     // qkv after 1x1 conv (later aliased)
  float* ws2 = ws1 + qkvElems;                   // qkv after 3x3 conv
  float* wp2 = ws2 + qkvElems;                   // repacked 3x3 weights [9][384][384]
  float* w1t = wp2 + (size_t)9 * C3 * C3;        // repacked qkv1 weights [128][384]
  float* wpt = w1t + (size_t)CDIM * C3;          // repacked proj weights [128][128]
  float* wsA = ws1;                              // attention output aliases ws1 (dead)

  // Repack weights (deterministic, every call).
  {
    const int t2 = C3 * C3 * 9, t1 = C3 * CDIM, tp = CDIM * CDIM;
    repack_kernel<<<(t2 + 255) / 256, 256, 0, stream>>>(qkv2_w, wp2, C3, C3, 9);
    repack_kernel<<<(t1 + 255) / 256, 256, 0, stream>>>(qkv1_w, w1t, C3, CDIM, 1);
    repack_kernel<<<(tp + 255) / 256, 256, 0, stream>>>(proj_w, wpt, CDIM, CDIM, 1);
  }

  dim3 blk(32);
  qkv1_kernel<<<dim3(PP / 16, C3 / 32, 4), blk, 0, stream>>>(x, w1t, ws1);
  qkv2_kernel<<<dim3((HH * WW) / (32 * 32) * 8, C3 / 32, 4), blk, 0, stream>>>(ws1, wp2, ws2);

  const size_t shbytes = (size_t)3 * 128 * LP * sizeof(float);  // ~203 KB LDS
  attn_kernel<<<dim3(CDIM, 4), dim3(256), shbytes, stream>>>(ws2, scale, wsA);

  proj_kernel<<<dim3(PP / 16, CDIM / 32, 4), blk, 0, stream>>>(wsA, proj_w != nullptr ? wpt : wpt, out);
}